// NormalizationNoise_7722351198658
// MI455X (gfx1250) — compile-verified
//
#include <hip/hip_runtime.h>
#include <hip/hip_bf16.h>
#include <math.h>

// Tensor: B=32, H=128, W=128, C=64  -> N = 33,554,432 f32 elements
// Quad (float4) view: Q = N/4 = 8,388,608. Quad index fields:
//   c4 = q[3:0] (16 quads per C-row), w = q[10:4], h = q[17:11], b = q[22:18]
#define NN_Q   (1 << 23)
#define NN_NF  33554432.0f
#define NN_K   1048576.0f

typedef __attribute__((ext_vector_type(2))) float v2f;
typedef __attribute__((ext_vector_type(4))) float f32x4;
typedef __attribute__((ext_vector_type(8))) float v8f;

#if defined(__has_builtin)
#if __has_builtin(__builtin_amdgcn_wmma_f32_16x16x4_f32)
#define NN_HAVE_WMMA_F32X4 1
#endif
#endif

// ---- wave32 reduction: matrix-pipe assisted ------------------------------
// A (16x4 f32, 2 VGPRs/lane) holds {s, 0}; B = all-ones. D[m,n] = s[m]+s[m+16]
// for every n, independent of the K slot ordering. Lane n (n<16) then sums its
// 8 D VGPRs (rows 0..7), lane n+16 sums rows 8..15; one xor-16 shuffle adds the
// halves -> full wave sum in every lane. Compiler inserts WMMA hazard NOPs.
__device__ __forceinline__ float nn_wave_reduce(float s) {
#ifdef NN_HAVE_WMMA_F32X4
  v2f a; a.x = s;    a.y = 0.0f;
  v2f b; b.x = 1.0f; b.y = 1.0f;
  v8f c = {};
  v8f d = __builtin_amdgcn_wmma_f32_16x16x4_f32(
      /*neg_a=*/false, a, /*neg_b=*/false, b,
      /*c_mod=*/(short)0, c, /*reuse_a=*/false, /*reuse_b=*/false);
  float t = d[0] + d[1] + d[2] + d[3] + d[4] + d[5] + d[6] + d[7];
  t += __shfl_xor(t, 16, 32);
  return t;
#else
  for (int m = 16; m >= 1; m >>= 1) s += __shfl_xor(s, m, 32);
  return s;
#endif
}

// ---- per-element math -----------------------------------------------------
// u+iv = conj(z_T)*z = (a*at+1) + i(at-a);  sqrt(u+iv) = x+iy (principal)
// z_norm  = 1024*(a+i)/(x+iy)  -> Re=(a*x+y)*1024/r, Im=(x-a*y)*1024/r, r=|u+iv|
// z_norm_T= 1024*(at+i)/(x-iy) -> Re=(at*x-y)*1024/r, Im=(at*y+x)*1024/r
__device__ __forceinline__ void nn_znorm(float a, float at,
                                         float& x, float& y, float& inv) {
  float u = fmaf(a, at, 1.0f);
  float v = at - a;
  float r = sqrtf(fmaf(u, u, v * v));          // r >= 1, division is safe
  x = sqrtf(0.5f * fmaxf(r + u, 0.0f));
  y = copysignf(sqrtf(0.5f * fmaxf(r - u, 0.0f)), v);
  inv = 1024.0f / r;
}

// ---- pass 1: dual reduction ----------------------------------------------
__global__ __launch_bounds__(256) void nn_reduce_kernel(
    const float* __restrict__ z, float* __restrict__ acc) {
  const f32x4* z4 = (const f32x4*)z;
  int tid = blockIdx.x * blockDim.x + threadIdx.x;
  int stride = gridDim.x * blockDim.x;
  float s1 = 0.0f, s2 = 0.0f;
  for (int q = tid; q < NN_Q; q += stride) {
    int c4 = q & 15, w = (q >> 4) & 127, h = (q >> 11) & 127, b = q >> 18;
    int qt = (b << 18) | (w << 11) | (h << 4) | c4;
    f32x4 av = z4[q];        // direct row, coalesced 256B
    f32x4 tv = z4[qt];       // transposed-partner row, also coalesced
#pragma unroll
    for (int j = 0; j < 4; ++j) {
      float a = av[j], at = tv[j];
      float x, y, inv;
      nn_znorm(a, at, x, y, inv);
      float znr  = fmaf(a, x, y)    * inv;
      float zni  = fmaf(-a, y, x)   * inv;
      float znrt = fmaf(at, x, -y)  * inv;
      float znit = fmaf(at, y, x)   * inv;
      s1 = fmaf(znrt, znr, s1);  s1 = fmaf(znit, zni, s1);  // Re(conj(znT)*zn)
      s2 = fmaf(znr, znr, s2);   s2 = fmaf(zni, zni, s2);   // |zn|^2
    }
  }
  // no divergence so far -> EXEC all-ones for the WMMA reduce
  float w1 = nn_wave_reduce(s1);
  float w2 = nn_wave_reduce(s2);
  __shared__ float l1[8], l2[8];
  int wid = threadIdx.x >> 5, lane = threadIdx.x & 31;
  if (lane == 0) { l1[wid] = w1; l2[wid] = w2; }
  __syncthreads();
  if (threadIdx.x == 0) {
    float t1 = 0.0f, t2 = 0.0f;
#pragma unroll
    for (int i = 0; i < 8; ++i) { t1 += l1[i]; t2 += l2[i]; }
    atomicAdd(&acc[0], t1);
    atomicAdd(&acc[1], t2);
  }
}

// ---- pass 2: scalar finalize ---------------------------------------------
__global__ void nn_finalize_kernel(float* __restrict__ acc) {
  float S1 = acc[0], S2 = acc[1];
  float pwr = S1 / NN_NF;                       // mean over all 32M elements
  float coefA, coefB, sumZ2;
  if (pwr > 1.0f) {                             // Z = z_norm*(pwr-i)/(pwr^2+1)
    float denom = fmaf(pwr, pwr, 1.0f);
    coefA = pwr / denom;
    coefB = 1.0f / denom;
    sumZ2 = S2 / denom;                         // sum|Z|^2 = sum|zn|^2/(pwr^2+1)
  } else {
    coefA = 1.0f; coefB = 0.0f; sumZ2 = S2;
  }
  float sig_pwr = sumZ2 / NN_K;
  float sigma = sqrtf(sig_pwr / 200.0f);        // /snr(=100)/2
  acc[4] = coefA; acc[5] = coefB; acc[6] = sigma;
}

// ---- pass 3: recompute z_norm, rescale, add noise ------------------------
__global__ __launch_bounds__(256) void nn_awgn_kernel(
    const float* __restrict__ z, const float* __restrict__ rnd,
    const float* __restrict__ acc, float* __restrict__ out) {
  const f32x4* z4 = (const f32x4*)z;
  const f32x4* r4 = (const f32x4*)rnd;
  f32x4* o4 = (f32x4*)out;
  float coefA = acc[4], coefB = acc[5], sigma = acc[6];
  int tid = blockIdx.x * blockDim.x + threadIdx.x;
  int stride = gridDim.x * blockDim.x;
  for (int q = tid; q < NN_Q; q += stride) {
    int c4 = q & 15, w = (q >> 4) & 127, h = (q >> 11) & 127, b = q >> 18;
    int qt = (b << 18) | (w << 11) | (h << 4) | c4;
    f32x4 av = z4[q];                                   // hot in L2 from pass 1
    f32x4 tv = z4[qt];
    f32x4 rv = __builtin_nontemporal_load(&r4[q]);      // single-use: NT
    f32x4 ov;
#pragma unroll
    for (int j = 0; j < 4; ++j) {
      float a = av[j], at = tv[j];
      float x, y, inv;
      nn_znorm(a, at, x, y, inv);
      float znr = fmaf(a, x, y)  * inv;
      float zni = fmaf(-a, y, x) * inv;
      ov[j] = fmaf(sigma, rv[j], fmaf(coefA, znr, coefB * zni));
    }
    __builtin_nontemporal_store(ov, &o4[q]);            // single-use: NT
  }
}

extern "C" void kernel_launch(void* const* d_in, const int* in_sizes, int n_in,
                              void* d_out, int out_size, void* d_ws, size_t ws_size,
                              hipStream_t stream) {
  const float* z   = (const float*)d_in[0];   // z_tilta, 33.5M f32
  const float* rnd = (const float*)d_in[1];   // rand_dist, 33.5M f32
  float* out = (float*)d_out;
  float* acc = (float*)d_ws;                  // [0]=S1 [1]=S2 [4..6]=coefs/sigma

  hipMemsetAsync(acc, 0, 2 * sizeof(float), stream);    // re-zero accumulators every call

  dim3 block(256);                            // 8 wave32 waves / block
  dim3 grid(2048);                            // 524288 threads, 16 quads each
  nn_reduce_kernel<<<grid, block, 0, stream>>>(z, acc);
  nn_finalize_kernel<<<1, 1, 0, stream>>>(acc);
  nn_awgn_kernel<<<grid, block, 0, stream>>>(z, rnd, acc, out);
}